// Radon_24592982737502
// MI455X (gfx1250) — compile-verified
//
#include <hip/hip_runtime.h>
#include <stdint.h>

// Radon transform for MI455X (gfx1250).
//   - Per-batch 256x256 image staged into WGP LDS by the CDNA5 Tensor Data
//     Mover with pad_enable: DMA writes a 258-stride padded layout whose
//     borders we zero, so bilinear taps need no boundary masks (clamp into
//     zero cells == padding_mode='zeros').
//   - wave32 / 1024-thread (32-wave) workgroups; threads walk rays
//     incrementally (pixel-space step along a ray is exactly (sin, cos)).

#define IMG   256
#define PAD   258        // IMG + 2 (1-cell zero border each side)
#define NA    512
#define NB    4
#define APW   8          // angles per workgroup
#define BLOCK 1024       // 256 w-positions x 4 angle slots
#define SLOTS 4          // BLOCK / IMG
#define PI_F  3.14159265358979323846f

typedef unsigned int v4u __attribute__((ext_vector_type(4)));
typedef int          v8i __attribute__((ext_vector_type(8)));
typedef int          v4i __attribute__((ext_vector_type(4)));

__global__ void __launch_bounds__(BLOCK)
radon_kernel(const float* __restrict__ img, float* __restrict__ out) {
    extern __shared__ float smem[];   // PAD*PAD floats = 266,256 B (< 320 KB WGP LDS)

    const int b      = blockIdx.y;
    const int a_base = blockIdx.x * APW;
    const int tid    = threadIdx.x;

    // ---- Zero the 1-cell border (top/bottom rows, left/right cols) ----
    if (tid < PAD) {
        smem[tid]                       = 0.0f;  // row y=-1
        smem[(PAD - 1) * PAD + tid]     = 0.0f;  // row y=IMG
        smem[tid * PAD]                 = 0.0f;  // col x=-1
        smem[tid * PAD + (PAD - 1)]     = 0.0f;  // col x=IMG
    }

    // ---- Stage image[b] Global -> LDS interior via the Tensor Data Mover ----
    // pad_enable: after every 256 dwords (one row) skip 2 dwords in LDS, so
    // rows land at stride PAD starting at (1,1). One wave issues (TDM is
    // wave-level; EXEC ignored).
    if (tid < 32) {
        const uint64_t gaddr = (uint64_t)(uintptr_t)(img + (size_t)b * IMG * IMG);
        const uint32_t ldsa  = (uint32_t)(uintptr_t)(void*)smem + (uint32_t)((PAD + 1) * sizeof(float));

        // D# group 0: count=1 (valid), lds_addr, global_addr[56:0], type=2
        v4u g0;
        g0.x = 1u;
        g0.y = ldsa;
        g0.z = (uint32_t)gaddr;
        g0.w = (uint32_t)((gaddr >> 32) & 0x01FFFFFFu) | (2u << 30);

        // D# group 1: data_size=4B; pad_enable, pad_interval=7 (256 dw),
        // pad_amount=1 (2 dw); tensor 256x256; tile 256x256; dim0 stride 256.
        v8i g1;
        g1[0] = (int)0x03D20000u;           // data_size=2 | pad_en | intv=7 | amt=1
        g1[1] = (int)((IMG & 0xFFFF) << 16);// atomic_bar_addr=0 | tensor_dim0.lo16
        g1[2] = (int)((IMG & 0xFFFF) << 16);// tensor_dim0.hi16=0 | tensor_dim1.lo16
        g1[3] = (int)(IMG << 16);           // tensor_dim1.hi16=0 | tile_dim0
        g1[4] = IMG;                        // tile_dim1 | tile_dim2=0
        g1[5] = IMG;                        // tensor_dim0_stride.lo32
        g1[6] = 0;                          // stride0.hi16 | stride1.lo16
        g1[7] = 0;                          // stride1.hi32

        const v4i z4 = {0, 0, 0, 0};        // 2-D tile: groups 2/3 unused
        const v8i z8 = {0, 0, 0, 0, 0, 0, 0, 0};
        __builtin_amdgcn_tensor_load_to_lds(g0, g1, z4, z4, z8, 0);
        __builtin_amdgcn_s_wait_tensorcnt(0);
    }
    __syncthreads();

    // ---- Compute: thread = (w, angle-slot); APW/SLOTS angles per thread ----
    const int   w     = tid & (IMG - 1);
    const int   slot  = tid >> 8;                       // 0..3
    const float X     = -1.0f + 2.0f * (float)w / (float)(IMG - 1);
    const float scale = 0.5f * (float)(IMG - 1);

#pragma unroll
    for (int rep = 0; rep < APW / SLOTS; ++rep) {
        const int   a     = a_base + slot + rep * SLOTS;
        const float theta = (float)a * (PI_F / (float)NA);
        float s, c;
        sincosf(theta, &s, &c);

        // Pixel-space start of the ray at h=0 (Y=-1); per-h step is (s, c).
        const float fx0 = ( c * X - s + 1.0f) * scale;
        const float fy0 = (-s * X - c + 1.0f) * scale;

        float acc = 0.0f;
        float hf  = 0.0f;
#pragma unroll 4
        for (int h = 0; h < IMG; ++h) {
            const float fx = fmaf(hf, s, fx0);
            const float fy = fmaf(hf, c, fy0);
            hf += 1.0f;

            const int   x0  = __float2int_rd(fx);       // floor
            const int   y0  = __float2int_rd(fy);
            const float wx1 = fx - (float)x0;
            const float wy1 = fy - (float)y0;
            const float wx0 = 1.0f - wx1;
            const float wy0 = 1.0f - wy1;

            // Clamp into the padded array; border cells are zero, so any
            // out-of-range tap contributes 0 regardless of its weight.
            const int cx0 = min(max(x0 + 1, 0), PAD - 1);
            const int cx1 = min(max(x0 + 2, 0), PAD - 1);
            const int cy0 = min(max(y0 + 1, 0), PAD - 1);
            const int cy1 = min(max(y0 + 2, 0), PAD - 1);

            const float* r0 = smem + cy0 * PAD;
            const float* r1 = smem + cy1 * PAD;
            const float v00 = r0[cx0], v01 = r0[cx1];
            const float v10 = r1[cx0], v11 = r1[cx1];

            acc = fmaf(wy0, fmaf(wx0, v00, wx1 * v01), acc);
            acc = fmaf(wy1, fmaf(wx0, v10, wx1 * v11), acc);
        }

        // out shape (B, 1, W, A)
        out[((size_t)b * IMG + w) * NA + a] = acc;
    }
}

extern "C" void kernel_launch(void* const* d_in, const int* in_sizes, int n_in,
                              void* d_out, int out_size, void* d_ws, size_t ws_size,
                              hipStream_t stream) {
    (void)in_sizes; (void)n_in; (void)d_ws; (void)ws_size; (void)out_size;
    const float* img = (const float*)d_in[0];
    float*       out = (float*)d_out;

    const size_t smem_bytes = (size_t)PAD * PAD * sizeof(float);  // ~260 KB
    (void)hipFuncSetAttribute((const void*)radon_kernel,
                              hipFuncAttributeMaxDynamicSharedMemorySize,
                              (int)smem_bytes);

    dim3 grid(NA / APW, NB);   // (64, 4)
    dim3 block(BLOCK);         // 32 wave32s
    radon_kernel<<<grid, block, smem_bytes, stream>>>(img, out);
}